// MultiHeadAttentionLayer_35940286333296
// MI455X (gfx1250) — compile-verified
//
#include <hip/hip_runtime.h>
#include <hip/hip_bf16.h>

typedef __attribute__((ext_vector_type(2))) float v2f;
typedef __attribute__((ext_vector_type(8))) float v8f;

constexpr int B_ = 4, S_ = 2048, E_ = 1024, H_ = 16, D_ = 64;   // H_*D_ == E_

// ---------------------------------------------------------------------------
// CDNA5 async global->LDS copy (16B per lane), tracked by ASYNCcnt.
// LDS offset = low 32 bits of the flat shared pointer (LDS aperture rule).
// ---------------------------------------------------------------------------
__device__ __forceinline__ void async_ld16(void* lds_ptr, const void* gptr) {
  const unsigned lds = (unsigned)(uintptr_t)lds_ptr;
  const unsigned long long ga = (unsigned long long)(uintptr_t)gptr;
  asm volatile("global_load_async_to_lds_b128 %0, %1, off"
               :: "v"(lds), "v"(ga)
               : "memory");
}
__device__ __forceinline__ void wait_async0() {
  asm volatile("s_wait_asynccnt 0" ::: "memory");
}

// ---------------------------------------------------------------------------
// LayerNorm: one block (256 threads) per row of 1024 elements.
// ---------------------------------------------------------------------------
__global__ __launch_bounds__(256) void ln_kernel(const float* __restrict__ x,
                                                 const float* __restrict__ gamma,
                                                 const float* __restrict__ beta,
                                                 float* __restrict__ xn) {
  const int row = blockIdx.x;
  const int t = threadIdx.x;
  const float4 val = ((const float4*)(x + (size_t)row * E_))[t];

  __shared__ float red[8];
  float s = val.x + val.y + val.z + val.w;
#pragma unroll
  for (int off = 16; off > 0; off >>= 1) s += __shfl_xor(s, off, 32);
  if ((t & 31) == 0) red[t >> 5] = s;
  __syncthreads();
  float mu = 0.f;
#pragma unroll
  for (int i = 0; i < 8; ++i) mu += red[i];
  mu *= (1.0f / (float)E_);
  __syncthreads();

  const float d0 = val.x - mu, d1 = val.y - mu, d2 = val.z - mu, d3 = val.w - mu;
  float s2 = d0 * d0 + d1 * d1 + d2 * d2 + d3 * d3;
#pragma unroll
  for (int off = 16; off > 0; off >>= 1) s2 += __shfl_xor(s2, off, 32);
  if ((t & 31) == 0) red[t >> 5] = s2;
  __syncthreads();
  float var = 0.f;
#pragma unroll
  for (int i = 0; i < 8; ++i) var += red[i];
  var *= (1.0f / (float)E_);
  const float rstd = rsqrtf(var + 1e-5f);

  const float4 gv = ((const float4*)gamma)[t];
  const float4 bv = ((const float4*)beta)[t];
  float4 o;
  o.x = d0 * rstd * gv.x + bv.x;
  o.y = d1 * rstd * gv.y + bv.y;
  o.z = d2 * rstd * gv.z + bv.z;
  o.w = d3 * rstd * gv.w + bv.w;
  ((float4*)(xn + (size_t)row * E_))[t] = o;
}

// ---------------------------------------------------------------------------
// fp32 WMMA GEMM: C[M,N] = A[M,K] * W[K,N] + bias[N] (+ residual)
// 128 threads = 4 waves; block computes 64x64; wave w does rows [16w,16w+16).
// Double-buffered LDS tiles filled by async global->LDS DMA; tile i+1 loads
// overlap tile i WMMAs. Pads (36/68) keep frag reads bank-conflict-free.
// ---------------------------------------------------------------------------
__global__ __launch_bounds__(128) void gemm_wmma_f32(const float* __restrict__ A,
                                                     const float* __restrict__ W,
                                                     const float* __restrict__ bias,
                                                     const float* __restrict__ residual,
                                                     float* __restrict__ C,
                                                     int M, int N, int K) {
  const int n0 = blockIdx.x * 64;
  const int m0 = blockIdx.y * 64;
  const int t = threadIdx.x;
  const int wave = t >> 5;
  const int lane = t & 31;
  const int lhalf = lane >> 4;
  const int l16 = lane & 15;

  __shared__ float lA[2][64 * 36];  // 64 rows x 32 k, stride 36
  __shared__ float lB[2][32 * 68];  // 32 k x 64 cols, stride 68

  v8f acc[4] = {v8f{}, v8f{}, v8f{}, v8f{}};

  auto stage = [&](int buf, int k0) {
#pragma unroll
    for (int i = 0; i < 4; ++i) {  // A: 512 16B chunks / block
      const int c = t + i * 128;
      const int r = c >> 3, cc = (c & 7) << 2;
      async_ld16(&lA[buf][r * 36 + cc], A + (size_t)(m0 + r) * K + k0 + cc);
    }
#pragma unroll
    for (int i = 0; i < 4; ++i) {  // B: 512 16B chunks / block
      const int c = t + i * 128;
      const int r = c >> 4, cc = (c & 15) << 2;
      async_ld16(&lB[buf][r * 68 + cc], W + (size_t)(k0 + r) * N + n0 + cc);
    }
  };

  stage(0, 0);

  for (int k0 = 0; k0 < K; k0 += 32) {
    const int buf = (k0 >> 5) & 1;
    wait_async0();          // this wave's pending tile is resident
    __syncthreads();        // every wave's chunks are resident / prev reads done
    if (k0 + 32 < K) stage(buf ^ 1, k0 + 32);  // prefetch under compute

#pragma unroll
    for (int kk = 0; kk < 32; kk += 4) {
      // A frag: lanes 0-15 hold k=kk,kk+1 ; lanes 16-31 hold k=kk+2,kk+3
      const v2f a = *(const v2f*)(&lA[buf][(wave * 16 + l16) * 36 + kk + 2 * lhalf]);
#pragma unroll
      for (int nt = 0; nt < 4; ++nt) {
        v2f b;
        b.x = lB[buf][(kk + 2 * lhalf + 0) * 68 + nt * 16 + l16];
        b.y = lB[buf][(kk + 2 * lhalf + 1) * 68 + nt * 16 + l16];
        acc[nt] = __builtin_amdgcn_wmma_f32_16x16x4_f32(
            false, a, false, b, (short)0, acc[nt], false, false);
      }
    }
  }

#pragma unroll
  for (int nt = 0; nt < 4; ++nt) {
#pragma unroll
    for (int r = 0; r < 8; ++r) {
      const int row = m0 + wave * 16 + r + 8 * lhalf;
      const int col = n0 + nt * 16 + l16;
      float v = acc[nt][r] + bias[col];
      if (residual) v += residual[(size_t)row * N + col];
      C[(size_t)row * N + col] = v;
    }
  }
}

// ---------------------------------------------------------------------------
// Causal flash attention, fp32 WMMA. Block = (qblock 64 rows, head, batch).
// q,k,v,o layout: [(b*S+s)*1024 + h*64 + d]. K/V tiles double-buffered via
// async global->LDS DMA.
// ---------------------------------------------------------------------------
__global__ __launch_bounds__(128) void attn_kernel(const float* __restrict__ q,
                                                   const float* __restrict__ k,
                                                   const float* __restrict__ v,
                                                   float* __restrict__ o) {
  const int q0 = blockIdx.x * 64;
  const int h = blockIdx.y;
  const int b = blockIdx.z;
  const int t = threadIdx.x;
  const int wave = t >> 5;
  const int lane = t & 31;
  const int lhalf = lane >> 4;
  const int l16 = lane & 15;
  const float scale = 0.125f;  // 1/sqrt(64)
  const float NEG_INF = -__builtin_inff();

  __shared__ float lK[2][64 * 68];
  __shared__ float lV[2][64 * 68];
  __shared__ float lP[4][16 * 17];

  const size_t bh = (size_t)b * S_ * E_ + (size_t)h * D_;

  auto stageKV = [&](int buf, int j0) {
#pragma unroll
    for (int i = 0; i < 8; ++i) {  // 64x64 K and V tiles, 16B chunks
      const int c = t + i * 128;
      const int r = c >> 4, cc = (c & 15) << 2;
      const size_t g = bh + (size_t)(j0 + r) * E_ + cc;
      async_ld16(&lK[buf][r * 68 + cc], k + g);
      async_ld16(&lV[buf][r * 68 + cc], v + g);
    }
  };

  // Q stripe for this wave, pre-packed in WMMA A-layout fragments.
  v2f qa[16];
  {
    const float* qrow = q + bh + (size_t)(q0 + wave * 16 + l16) * E_;
#pragma unroll
    for (int j = 0; j < 16; ++j) qa[j] = *(const v2f*)(qrow + 4 * j + 2 * lhalf);
  }

  v8f oacc[4] = {v8f{}, v8f{}, v8f{}, v8f{}};
  float mrun[8], lrun[8];
#pragma unroll
  for (int r = 0; r < 8; ++r) { mrun[r] = NEG_INF; lrun[r] = 0.f; }

  const int myq = q0 + wave * 16;

  stageKV(0, 0);

  for (int j0 = 0; j0 < q0 + 64; j0 += 64) {
    const int buf = (j0 >> 6) & 1;
    wait_async0();
    __syncthreads();
    if (j0 + 64 < q0 + 64) stageKV(buf ^ 1, j0 + 64);  // prefetch under compute

#pragma unroll
    for (int jj = 0; jj < 4; ++jj) {
      const int kbase = j0 + jj * 16;
      if (kbase > myq + 15) continue;  // wave-uniform: EXEC stays all-ones

      // S = Q * K^T
      v8f s = {};
#pragma unroll
      for (int kk4 = 0; kk4 < 16; ++kk4) {
        const int d = 4 * kk4 + 2 * lhalf;
        const v2f bfrag = *(const v2f*)(&lK[buf][(jj * 16 + l16) * 68 + d]);
        s = __builtin_amdgcn_wmma_f32_16x16x4_f32(
            false, qa[kk4], false, bfrag, (short)0, s, false, false);
      }

      // scale + causal mask + online softmax
#pragma unroll
      for (int r = 0; r < 8; ++r) {
        float sv = s[r] * scale;
        const int qi = myq + r + 8 * lhalf;
        const int kj = kbase + l16;
        if (kj > qi) sv = NEG_INF;

        float m = sv;
#pragma unroll
        for (int off = 8; off > 0; off >>= 1) m = fmaxf(m, __shfl_xor(m, off, 16));
        const float nm = fmaxf(mrun[r], m);
        const float pr = __expf(sv - nm);
        float psum = pr;
#pragma unroll
        for (int off = 8; off > 0; off >>= 1) psum += __shfl_xor(psum, off, 16);
        const float sc = __expf(mrun[r] - nm);
        lrun[r] = lrun[r] * sc + psum;
        mrun[r] = nm;
#pragma unroll
        for (int nt = 0; nt < 4; ++nt) oacc[nt][r] *= sc;
        // C-layout -> LDS (read back below in A-layout). Same-wave DS ops are
        // in-order; no block barrier needed, lP[wave] is wave-private.
        lP[wave][(r + 8 * lhalf) * 17 + l16] = pr;
      }

      // O += P * V
#pragma unroll
      for (int kk = 0; kk < 16; kk += 4) {
        v2f pa;
        pa.x = lP[wave][l16 * 17 + kk + 2 * lhalf + 0];
        pa.y = lP[wave][l16 * 17 + kk + 2 * lhalf + 1];
#pragma unroll
        for (int nt = 0; nt < 4; ++nt) {
          v2f bf;
          bf.x = lV[buf][(jj * 16 + kk + 2 * lhalf + 0) * 68 + nt * 16 + l16];
          bf.y = lV[buf][(jj * 16 + kk + 2 * lhalf + 1) * 68 + nt * 16 + l16];
          oacc[nt] = __builtin_amdgcn_wmma_f32_16x16x4_f32(
              false, pa, false, bf, (short)0, oacc[nt], false, false);
        }
      }
    }
  }

  // normalize + store
#pragma unroll
  for (int r = 0; r < 8; ++r) {
    const float inv = 1.0f / lrun[r];
    const size_t row = bh + (size_t)(myq + r + 8 * lhalf) * E_;
#pragma unroll
    for (int nt = 0; nt < 4; ++nt)
      o[row + nt * 16 + l16] = oacc[nt][r] * inv;
  }
}

// ---------------------------------------------------------------------------
extern "C" void kernel_launch(void* const* d_in, const int* in_sizes, int n_in,
                              void* d_out, int out_size, void* d_ws, size_t ws_size,
                              hipStream_t stream) {
  const float* x     = (const float*)d_in[0];
  const float* gamma = (const float*)d_in[1];
  const float* beta  = (const float*)d_in[2];
  const float* Wq    = (const float*)d_in[3];
  const float* bq    = (const float*)d_in[4];
  const float* Wk    = (const float*)d_in[5];
  const float* bk    = (const float*)d_in[6];
  const float* Wv    = (const float*)d_in[7];
  const float* bv    = (const float*)d_in[8];
  const float* Wo    = (const float*)d_in[9];
  const float* bo    = (const float*)d_in[10];
  float* out = (float*)d_out;

  const size_t rows = (size_t)B_ * S_;      // 8192
  const size_t sz = rows * E_;              // floats per tensor
  float* ws  = (float*)d_ws;
  float* xn  = ws;
  float* qb  = ws + sz;
  float* kb  = ws + 2 * sz;
  float* vb  = ws + 3 * sz;
  float* att = ws + 4 * sz;

  ln_kernel<<<dim3((unsigned)rows), 256, 0, stream>>>(x, gamma, beta, xn);

  const dim3 gg(E_ / 64, (unsigned)(rows / 64));
  gemm_wmma_f32<<<gg, 128, 0, stream>>>(xn, Wq, bq, nullptr, qb, (int)rows, E_, E_);
  gemm_wmma_f32<<<gg, 128, 0, stream>>>(xn, Wk, bk, nullptr, kb, (int)rows, E_, E_);
  gemm_wmma_f32<<<gg, 128, 0, stream>>>(xn, Wv, bv, nullptr, vb, (int)rows, E_, E_);

  attn_kernel<<<dim3(S_ / 64, H_, B_), 128, 0, stream>>>(qb, kb, vb, att);

  gemm_wmma_f32<<<gg, 128, 0, stream>>>(att, Wo, bo, x, out, (int)rows, E_, E_);
}